// ModelNew_5909874999910
// MI455X (gfx1250) — compile-verified
//
#include <hip/hip_runtime.h>

typedef __attribute__((ext_vector_type(2))) float v2f;
typedef __attribute__((ext_vector_type(8))) float v8f;

#define NDIM 4096
#define MTM  256      // macro tile rows per block (8 waves x 32 rows)
#define MTN  128      // macro tile cols per block
#define KT   16       // k-step staged through LDS
#define ASTR 20       // As row stride: 16B-aligned rows, conflict-free b64 frag reads
#define BW   144      // BsP width: 2*144 % 64 == 32 -> lane halves on disjoint banks

// C = triu(A @ B), A/B fp32 upper-triangular NDIMxNDIM, row-major.
__global__ __launch_bounds__(256) void triu_gemm_wmma_f32(
    const float* __restrict__ A,
    const float* __restrict__ B,
    float* __restrict__ C)
{
    const int br = blockIdx.y;          // macro-tile row (256 rows)
    const int bc = blockIdx.x;          // macro-tile col (128 cols)
    const int r0 = br * MTM;
    const int c0 = bc * MTN;
    const int tid = threadIdx.x;

    if (r0 >= c0 + MTN) {
        // Entire macro tile strictly below the diagonal: exact zero.
        // Must write it because the harness poisons d_out.
        for (int i = tid; i < (MTM * MTN) / 4; i += 256) {
            const int row  = i >> 5;          // MTN/4 == 32 float4 per row
            const int col4 = (i & 31) << 2;
            *(float4*)&C[(size_t)(r0 + row) * NDIM + (size_t)(c0 + col4)] =
                make_float4(0.f, 0.f, 0.f, 0.f);
        }
        return;
    }

    __shared__ float As[MTM][ASTR];         // A[r0+r][k0+kk], stride 20
    __shared__ float BsP[KT / 2][BW][2];    // BsP[kq][n][j] = B[k0+2kq+j][c0+n]

    const int lane = tid & 31;
    const int wave = tid >> 5;          // 0..7 -> 32-row strip within macro tile
    const int m    = lane & 15;         // 0..15
    const int h    = lane >> 4;         // 0..1  (lane half: K pair select / M+8)

    // 2 M-tiles x 8 N-tiles of 16x16 f32 accumulators (128 VGPRs)
    v8f acc[2][8] = {};

    // Triangular pruning of the K range:
    //   A upper-tri -> k >= r0 ; B upper-tri -> k < c0 + MTN
    const int kLo = r0;
    const int kHi = c0 + MTN;

    for (int k0 = kLo; k0 < kHi; k0 += KT) {
        // ---- stage A (MTM x KT) as float4: 4 lanes cover one 16-float row ----
        {
            const int ar  = tid >> 2;         // 0..63
            const int ac4 = (tid & 3) << 2;   // 0,4,8,12
            #pragma unroll
            for (int rr = 0; rr < 4; ++rr) {
                const int row = ar + rr * 64;
                *(float4*)&As[row][ac4] =
                    *(const float4*)&A[(size_t)(r0 + row) * NDIM + (size_t)(k0 + ac4)];
            }
        }
        // ---- stage B (KT x MTN) pair-interleaved over K ----
        {
            const int bcol = tid & (MTN - 1);
            const int brow = tid >> 7;        // 0..1
            #pragma unroll
            for (int rr = 0; rr < 8; ++rr)
                BsP[rr][bcol][brow] =         // row = brow + 2*rr
                    B[(size_t)(k0 + brow + 2 * rr) * NDIM + (size_t)(c0 + bcol)];
        }
        __syncthreads();

        // ---- compute: 4 k-substeps x (2 M-tiles x 8 N-tiles) = 64 WMMAs ----
        #pragma unroll
        for (int kk = 0; kk < KT; kk += 4) {
            // fp32 A fragment 16x4: lane(h,m): VGPR j holds A[M=m, K=kk+2h+j]
            // Pair is contiguous & 8B-aligned in As -> single ds_load_b64.
            const v2f a0 = *(const v2f*)&As[wave * 32 + m     ][kk + 2 * h];
            const v2f a1 = *(const v2f*)&As[wave * 32 + 16 + m][kk + 2 * h];
            const int kq = (kk >> 1) + h;     // pair row: K = 2*kq + j
            #pragma unroll
            for (int nt = 0; nt < 8; ++nt) {
                // fp32 B fragment 4x16: VGPR j holds B[K=kk+2h+j, N=m]
                const v2f b = *(const v2f*)&BsP[kq][nt * 16 + m][0];
                acc[0][nt] = __builtin_amdgcn_wmma_f32_16x16x4_f32(
                    false, a0, false, b, (short)0, acc[0][nt], false, false);
                acc[1][nt] = __builtin_amdgcn_wmma_f32_16x16x4_f32(
                    false, a1, false, b, (short)0, acc[1][nt], false, false);
            }
        }
        __syncthreads();
    }

    // ---- store with triu mask (covers diagonal-crossing blocks) ----
    // C/D layout: VGPR r, lane(h,m) -> element (M = r + 8h, N = m)
    #pragma unroll
    for (int s = 0; s < 2; ++s) {
        const int i0 = r0 + wave * 32 + s * 16;
        #pragma unroll
        for (int nt = 0; nt < 8; ++nt) {
            const int col = c0 + nt * 16 + m;
            #pragma unroll
            for (int r = 0; r < 8; ++r) {
                const int row = i0 + r + 8 * h;
                const float v = acc[s][nt][r];
                C[(size_t)row * NDIM + (size_t)col] = (row <= col) ? v : 0.0f;
            }
        }
    }
}

extern "C" void kernel_launch(void* const* d_in, const int* in_sizes, int n_in,
                              void* d_out, int out_size, void* d_ws, size_t ws_size,
                              hipStream_t stream) {
    (void)in_sizes; (void)n_in; (void)out_size; (void)d_ws; (void)ws_size;
    const float* A = (const float*)d_in[0];
    const float* B = (const float*)d_in[1];
    float* C = (float*)d_out;

    dim3 grid(NDIM / MTN, NDIM / MTM);  // 32 x 16 macro tiles
    dim3 block(256);                    // 8 wave32s
    triu_gemm_wmma_f32<<<grid, block, 0, stream>>>(A, B, C);
}